// GCN_network_87986700026138
// MI455X (gfx1250) — compile-verified
//
#include <hip/hip_runtime.h>
#include <hip/hip_bf16.h>

typedef __attribute__((ext_vector_type(2))) float v2f;
typedef __attribute__((ext_vector_type(8))) float v8f;

__device__ __forceinline__ float lrelu(float v, float s) { return v >= 0.f ? v : s * v; }

// ---------------- element-wise / graph kernels ----------------

__global__ void fill_kernel(float* p, float v, int n) {
  int i = blockIdx.x * blockDim.x + threadIdx.x;
  if (i < n) p[i] = v;
}

__global__ void edge_deg_kernel(const int* __restrict__ dst, float* deg, int E) {
  int i = blockIdx.x * blockDim.x + threadIdx.x;
  if (i < E) atomicAdd(&deg[dst[i]], 1.0f);
}

__global__ void dinv_kernel(float* dinv, int n) {
  int i = blockIdx.x * blockDim.x + threadIdx.x;
  if (i < n) dinv[i] = rsqrtf(dinv[i] + 1.0f);   // +1 = self loop
}

// y[i] = dinv[i]^2 * x[i]  (self-loop term; also fully initializes y)
__global__ void prop_self_kernel(const float* __restrict__ x, const float* __restrict__ dinv,
                                 float* __restrict__ y, int n) {
  int idx = blockIdx.x * blockDim.x + threadIdx.x;
  if (idx < n * 128) {
    float d = dinv[idx >> 7];
    y[idx] = d * d * x[idx];
  }
}

// one wave32 per edge; lane handles a float4 of the 128-wide feature row
__global__ void prop_edge_kernel(const float* __restrict__ x, const int* __restrict__ src,
                                 const int* __restrict__ dst, const float* __restrict__ dinv,
                                 float* __restrict__ y, int E) {
  int t = blockIdx.x * blockDim.x + threadIdx.x;
  int e = __builtin_amdgcn_readfirstlane(t >> 5);   // uniform -> scalar loads below
  if (e >= E) return;
  int lane = t & 31;
  int s = src[e], d = dst[e];
  float w = dinv[s] * dinv[d];
  float4 xv = *((const float4*)(x + (size_t)s * 128) + lane);
  float* yp = y + (size_t)d * 128 + lane * 4;
  atomicAdd(yp + 0, w * xv.x);
  atomicAdd(yp + 1, w * xv.y);
  atomicAdd(yp + 2, w * xv.z);
  atomicAdd(yp + 3, w * xv.w);
}

// ---------------- WMMA fp32 GEMM: C[M x Ncol] = act(A[M x K] @ B[K x Ncol] + bias) ----------------
// block = 128 threads = 4 waves; each wave owns one 16x16 tile. M, K multiples of 16/8.
__global__ void gemm_wmma_kernel(const float* __restrict__ A, const float* __restrict__ B,
                                 const float* __restrict__ bias, float* __restrict__ C,
                                 int M, int K, int Ncol, float slope, int act) {
  extern __shared__ float smem[];
  const int rowBase = blockIdx.x * 16;
  const int tid = threadIdx.x;

  // A tile: rows rowBase..rowBase+15 are a contiguous 16*K float block -> flat float4 copy
  {
    const float4* A4 = (const float4*)(A + (size_t)rowBase * K);
    float4* S4 = (float4*)smem;
    const int tot = (16 * K) >> 2;
    for (int i = tid; i < tot; i += blockDim.x) S4[i] = A4[i];
  }
  __syncthreads();

  const int wave = tid >> 5, lane = tid & 31;
  const int nBase = (blockIdx.y * 4 + wave) * 16;
  if (nBase >= Ncol) return;                   // uniform per wave -> EXEC all-1s for WMMA

  const int mrow = lane & 15;                  // A: both lane halves index M=0..15
  const int g2 = (lane >> 4) * 2;              // lane half selects K pair {0,1} vs {2,3}
  const int col = nBase + (lane & 15);
  const float* lA = smem + mrow * K + g2;
  const int bstep = Ncol * 8;

  v8f acc0 = {0.f, 0.f, 0.f, 0.f, 0.f, 0.f, 0.f, 0.f};
  v8f acc1 = {0.f, 0.f, 0.f, 0.f, 0.f, 0.f, 0.f, 0.f};

  if (nBase + 16 <= Ncol) {                    // clean path: no bounds masking
    const float* Bp = B + g2 * Ncol + col;
    #pragma unroll 2
    for (int k = 0; k < K; k += 8) {           // two independent accumulator chains
      v2f a0, b0, a1, b1;
      a0.x = lA[k];     a0.y = lA[k + 1];
      a1.x = lA[k + 4]; a1.y = lA[k + 5];
      b0.x = Bp[0];          b0.y = Bp[Ncol];
      b1.x = Bp[Ncol * 4];   b1.y = Bp[Ncol * 5];
      Bp += bstep;
      acc0 = __builtin_amdgcn_wmma_f32_16x16x4_f32(false, a0, false, b0, (short)0, acc0, false, false);
      acc1 = __builtin_amdgcn_wmma_f32_16x16x4_f32(false, a1, false, b1, (short)0, acc1, false, false);
    }
    const float bv = bias ? bias[col] : 0.f;
    float* Cp = C + (rowBase + ((lane >> 4) * 8)) * Ncol + col;
    #pragma unroll
    for (int r = 0; r < 8; ++r) {
      float v = acc0[r] + acc1[r] + bv;
      if (act) v = lrelu(v, slope);
      Cp[r * Ncol] = v;
    }
  } else {                                     // tail path (only Ncol=8 final layer)
    const bool ok = col < Ncol;
    const int colc = ok ? col : 0;
    const float cmask = ok ? 1.f : 0.f;
    const float* Bp = B + g2 * Ncol + colc;
    for (int k = 0; k < K; k += 8) {
      v2f a0, b0, a1, b1;
      a0.x = lA[k];     a0.y = lA[k + 1];
      a1.x = lA[k + 4]; a1.y = lA[k + 5];
      b0.x = Bp[0] * cmask;        b0.y = Bp[Ncol] * cmask;
      b1.x = Bp[Ncol * 4] * cmask; b1.y = Bp[Ncol * 5] * cmask;
      Bp += bstep;
      acc0 = __builtin_amdgcn_wmma_f32_16x16x4_f32(false, a0, false, b0, (short)0, acc0, false, false);
      acc1 = __builtin_amdgcn_wmma_f32_16x16x4_f32(false, a1, false, b1, (short)0, acc1, false, false);
    }
    if (ok) {
      const float bv = bias ? bias[col] : 0.f;
      float* Cp = C + (rowBase + ((lane >> 4) * 8)) * Ncol + col;
      #pragma unroll
      for (int r = 0; r < 8; ++r) {
        float v = acc0[r] + acc1[r] + bv;
        if (act) v = lrelu(v, slope);
        Cp[r * Ncol] = v;
      }
    }
  }
}

// ---------------- GAT kernels ----------------

__device__ __forceinline__ void atomicMaxFloat(float* addr, float v) {
  if (v >= 0.f) atomicMax((int*)addr, __float_as_int(v));
  else          atomicMin((unsigned int*)addr, __float_as_uint(v));
}

__global__ void gat_coeff_kernel(const float* __restrict__ hg, const float* __restrict__ att_s,
                                 const float* __restrict__ att_d, float* a_s, float* a_d,
                                 float* m, int N) {
  int i = blockIdx.x * blockDim.x + threadIdx.x;
  if (i >= N) return;
  const float* h = hg + (size_t)i * 64;
  float as = 0.f, ad = 0.f;
  #pragma unroll 8
  for (int k = 0; k < 64; ++k) { float hv = h[k]; as += hv * att_s[k]; ad += hv * att_d[k]; }
  a_s[i] = as; a_d[i] = ad;
  m[i] = lrelu(as + ad, 0.2f);                 // self-loop edge initializes the running max
}

__global__ void gat_max_kernel(const int* __restrict__ src, const int* __restrict__ dst,
                               const float* __restrict__ a_s, const float* __restrict__ a_d,
                               float* m, int E) {
  int i = blockIdx.x * blockDim.x + threadIdx.x;
  if (i >= E) return;
  int s = src[i], d = dst[i];
  atomicMaxFloat(&m[d], lrelu(a_s[s] + a_d[d], 0.2f));
}

// initializes s[] and out[] with the self-loop contribution (no atomics needed)
__global__ void gat_init_kernel(const float* __restrict__ hg, const float* __restrict__ a_s,
                                const float* __restrict__ a_d, const float* __restrict__ m,
                                float* s, float* __restrict__ out, int N) {
  int idx = blockIdx.x * blockDim.x + threadIdx.x;
  if (idx >= N * 64) return;
  int i = idx >> 6, f = idx & 63;
  float e = lrelu(a_s[i] + a_d[i], 0.2f);
  float w = __expf(e - m[i]);
  if (f == 0) s[i] = w;
  out[idx] = w * hg[idx];
}

// one wave32 per edge; lane handles 2 of 64 features
__global__ void gat_edge_kernel(const int* __restrict__ src, const int* __restrict__ dst,
                                const float* __restrict__ a_s, const float* __restrict__ a_d,
                                const float* __restrict__ m, float* s,
                                const float* __restrict__ hg, float* __restrict__ out, int E) {
  int t = blockIdx.x * blockDim.x + threadIdx.x;
  int e = __builtin_amdgcn_readfirstlane(t >> 5);   // uniform -> scalar loads below
  if (e >= E) return;
  int lane = t & 31;
  int si = src[e], di = dst[e];
  float ea = lrelu(a_s[si] + a_d[di], 0.2f);
  float w = __expf(ea - m[di]);
  if (lane == 0) atomicAdd(&s[di], w);
  float2 hv = *((const float2*)(hg + (size_t)si * 64) + lane);
  float* op = out + (size_t)di * 64 + lane * 2;
  atomicAdd(op + 0, w * hv.x);
  atomicAdd(op + 1, w * hv.y);
}

__global__ void gat_final_kernel(const float* __restrict__ out, const float* __restrict__ s,
                                 const float* __restrict__ bg, float* __restrict__ h3, int N) {
  int idx = blockIdx.x * blockDim.x + threadIdx.x;
  if (idx >= N * 64) return;
  int i = idx >> 6, f = idx & 63;
  h3[idx] = lrelu(out[idx] / s[i] + bg[f], 0.1f);
}

// ---------------- launch ----------------

extern "C" void kernel_launch(void* const* d_in, const int* in_sizes, int n_in,
                              void* d_out, int out_size, void* d_ws, size_t ws_size,
                              hipStream_t stream) {
  const float* x       = (const float*)d_in[0];
  const int*   ei      = (const int*)d_in[1];
  const float* W1      = (const float*)d_in[2];
  const float* b1      = (const float*)d_in[3];
  const float* W2      = (const float*)d_in[4];
  const float* b2      = (const float*)d_in[5];
  const float* Wg      = (const float*)d_in[6];
  const float* att_src = (const float*)d_in[7];
  const float* att_dst = (const float*)d_in[8];
  const float* bg      = (const float*)d_in[9];
  const float* Wl1     = (const float*)d_in[10];
  const float* bl1     = (const float*)d_in[11];
  const float* Wl2     = (const float*)d_in[12];
  const float* bl2     = (const float*)d_in[13];
  const float* Wl3     = (const float*)d_in[14];
  const float* bl3     = (const float*)d_in[15];

  const int N = in_sizes[0] / 128;   // 100000
  const int E = in_sizes[1] / 2;     // 1600000
  const int* src = ei;
  const int* dst = ei + E;

  float* ws   = (float*)d_ws;
  float* dinv = ws;
  float* a_s  = ws + (size_t)N;
  float* a_d  = ws + (size_t)2 * N;
  float* mmax = ws + (size_t)3 * N;
  float* ssum = ws + (size_t)4 * N;
  float* bufA = ws + (size_t)5 * N;            // N x 128
  float* bufB = bufA + (size_t)N * 128;        // N x 128

  const int TPB = 256;
  auto cdiv = [](long a, long b) { return (unsigned)((a + b - 1) / b); };

  // degree -> dinv (self loops folded in via +1)
  fill_kernel<<<cdiv(N, TPB), TPB, 0, stream>>>(dinv, 0.f, N);
  edge_deg_kernel<<<cdiv(E, TPB), TPB, 0, stream>>>(dst, dinv, E);
  dinv_kernel<<<cdiv(N, TPB), TPB, 0, stream>>>(dinv, N);

  auto prop = [&](const float* in, float* out_) {
    prop_self_kernel<<<cdiv((long)N * 128, TPB), TPB, 0, stream>>>(in, dinv, out_, N);
    prop_edge_kernel<<<cdiv((long)E * 32, TPB), TPB, 0, stream>>>(in, src, dst, dinv, out_, E);
  };
  auto gemm = [&](const float* A, const float* B, const float* bias, float* C,
                  int M, int K, int Ncol, int act) {
    dim3 g(M / 16, cdiv(Ncol, 64));
    gemm_wmma_kernel<<<g, 128, (size_t)16 * K * sizeof(float), stream>>>(
        A, B, bias, C, M, K, Ncol, 0.1f, act);
  };

  // conv1: SGConv K=2 + lrelu(0.1)
  prop(x, bufA);
  prop(bufA, bufB);
  gemm(bufB, W1, b1, bufA, N, 128, 128, 1);
  // conv2: SGConv K=2 + lrelu(0.1)
  prop(bufA, bufB);
  prop(bufB, bufA);
  gemm(bufA, W2, b2, bufB, N, 128, 128, 1);
  // GAT: hg = h2 @ Wg (no bias, no act) -> bufA[N x 64]
  gemm(bufB, Wg, nullptr, bufA, N, 128, 64, 0);
  gat_coeff_kernel<<<cdiv(N, TPB), TPB, 0, stream>>>(bufA, att_src, att_dst, a_s, a_d, mmax, N);
  gat_max_kernel<<<cdiv(E, TPB), TPB, 0, stream>>>(src, dst, a_s, a_d, mmax, E);
  gat_init_kernel<<<cdiv((long)N * 64, TPB), TPB, 0, stream>>>(bufA, a_s, a_d, mmax, ssum, bufB, N);
  gat_edge_kernel<<<cdiv((long)E * 32, TPB), TPB, 0, stream>>>(src, dst, a_s, a_d, mmax, ssum,
                                                               bufA, bufB, E);
  float* h3 = bufB + (size_t)N * 64;           // N x 64, disjoint from out accumulator
  gat_final_kernel<<<cdiv((long)N * 64, TPB), TPB, 0, stream>>>(bufB, ssum, bg, h3, N);

  // MLP over [N/10, 640]
  const int G = N / 10;                        // 10000, multiple of 16
  gemm(h3, Wl1, bl1, bufA, G, 640, 512, 1);
  gemm(bufA, Wl2, bl2, bufB, G, 512, 256, 1);
  gemm(bufB, Wl3, bl3, (float*)d_out, G, 256, 8, 0);
}